// Neuron_41841571398457
// MI455X (gfx1250) — compile-verified
//
#include <hip/hip_runtime.h>

// Noisy-LIF scan, MI455X (gfx1250, wave32).
// x, noise: [B, T, N] fp32 with B=16, T=64, N=65536 (fixed by reference).
// out: [B, T, N] fp32 spike train.
// Pure HBM-streaming kernel: 805 MB traffic, ~34.6 us roofline @ 23.3 TB/s.

#define TAU_INV 0.5f
#define SIGMA   0.5f
#define THR     1.0f

// Clang native vector (NOT HIP_vector_type) so __builtin_nontemporal_* accepts it.
typedef __attribute__((ext_vector_type(4))) float v4f;

namespace {
constexpr int kB = 16;
constexpr int kT = 64;
constexpr int kN = 65536;
constexpr int kNVec = kN / 4;          // float4 lanes per (b,t) row = 16384 = 2^14
constexpr int kNVecLog2 = 14;
constexpr int kThreads = kB * kNVec;   // 262,144 threads = 8192 wave32s
constexpr int kBlock = 256;            // 8 wave32s per block
}

// One LIF step on a scalar lane; updates u in place, returns the spike (0/1).
__device__ __forceinline__ float lif_step(float& u, float xv, float nv) {
    // u = 0.5*u + x - 0.5*n  (two FMAs)
    float un = fmaf(TAU_INV, u, fmaf(-SIGMA, nv, xv));
    bool  sp = un > THR;               // H(u - 1) with strict >
    u = sp ? 0.0f : un;                // hard reset
    return sp ? 1.0f : 0.0f;
}

__global__ __launch_bounds__(kBlock)
void lif_scan_kernel(const v4f* __restrict__ x,
                     const v4f* __restrict__ nz,
                     v4f* __restrict__ o)
{
    const int tid = blockIdx.x * kBlock + threadIdx.x;   // [0, kThreads)
    const int n4  = tid & (kNVec - 1);                   // float4 index within row
    const int b   = tid >> kNVecLog2;                    // batch
    // Index in float4 units: b*T*nvec + t*nvec + n4. Max ~16.7M -> 32-bit safe.
    const int base = b * (kT * kNVec) + n4;

    v4f u = (v4f){0.0f, 0.0f, 0.0f, 0.0f};

    #pragma unroll 8
    for (int t = 0; t < kT; ++t) {
        const int idx = base + t * kNVec;
        // Streaming data, zero reuse, working set >> 192 MB L2:
        // non-temporal hints (gfx1250 TH field) on both loads and the store.
        v4f xv = __builtin_nontemporal_load(&x[idx]);
        v4f nv = __builtin_nontemporal_load(&nz[idx]);

        float ux = u.x, uy = u.y, uz = u.z, uw = u.w;
        v4f ov;
        ov.x = lif_step(ux, xv.x, nv.x);
        ov.y = lif_step(uy, xv.y, nv.y);
        ov.z = lif_step(uz, xv.z, nv.z);
        ov.w = lif_step(uw, xv.w, nv.w);
        u.x = ux; u.y = uy; u.z = uz; u.w = uw;

        __builtin_nontemporal_store(ov, &o[idx]);
    }
}

extern "C" void kernel_launch(void* const* d_in, const int* in_sizes, int n_in,
                              void* d_out, int out_size, void* d_ws, size_t ws_size,
                              hipStream_t stream) {
    (void)n_in; (void)d_ws; (void)ws_size; (void)out_size; (void)in_sizes;

    const v4f* x  = (const v4f*)d_in[0];   // [B,T,N] fp32
    const v4f* nz = (const v4f*)d_in[1];   // [B,T,N] fp32 unit noise
    v4f* o        = (v4f*)d_out;           // [B,T,N] fp32 spikes

    dim3 block(kBlock);
    dim3 grid(kThreads / kBlock);          // 1024 blocks
    lif_scan_kernel<<<grid, block, 0, stream>>>(x, nz, o);
}